// RGCNLayerWithSkip_28243704938827
// MI455X (gfx1250) — compile-verified
//
#include <hip/hip_runtime.h>
#include <hip/hip_bf16.h>
#include <stddef.h>

// Problem constants (from reference): N=50000, E=600000, D=128, R=8
#define NN    50000
#define EE    600000
#define DD    128
#define RR    8
#define NTIL  (NN / 16)          // 3125 row tiles, exact
#define LDSPITCH 130             // pad 16x128 fp32 tile to avoid LDS bank conflicts

typedef __attribute__((ext_vector_type(2))) float v2f;
typedef __attribute__((ext_vector_type(8))) float v8f;

// ---------------------------------------------------------------------------
// Kernel 1: h_rel[n, r, f] = sum_d h[n,d] * rel_weight[r,d,f]
// One block = one 16-node row tile. 8 waves; each wave owns col tiles
// ct = wave, wave+8, ... of the 64 (R*D/16) 16-wide output column tiles.
// V_WMMA_F32_16X16X4_F32, K loop of 32 steps covers K=128.
// ---------------------------------------------------------------------------
__global__ void __launch_bounds__(256)
k_hrel_wmma(const float* __restrict__ h,
            const float* __restrict__ relw,   // [R, D, D] row-major (r, d, f)
            float* __restrict__ hrel)         // [N, R, D]
{
    __shared__ float th[16 * LDSPITCH];

    const int row0 = blockIdx.x * 16;

    // Stage A tile (16 x 128) into LDS, coalesced.
    for (int i = threadIdx.x; i < 16 * 128; i += 256) {
        const int r = i >> 7;
        const int c = i & 127;
        th[r * LDSPITCH + c] = h[(size_t)(row0 + r) * DD + c];
    }
    __syncthreads();

    const int wave  = threadIdx.x >> 5;
    const int lane  = threadIdx.x & 31;
    const int m     = lane & 15;              // A row / D col within tile
    const int khalf = (lane >> 4) << 1;       // 0 or 2 (K sub-offset per lane half)
    const int rowh  = (lane >> 4) << 3;       // 0 or 8 (D row offset per lane half)

    for (int ct = wave; ct < (RR * DD) / 16; ct += 8) {
        const int rel  = ct >> 3;             // which relation
        const int ft   = ct & 7;              // which 16-wide f tile
        const int fcol = ft * 16 + m;
        const float* __restrict__ B = relw + (size_t)rel * DD * DD; // B[k][f]

        v8f acc = {};
        #pragma unroll 4
        for (int s = 0; s < 32; ++s) {
            const int kb = s * 4 + khalf;
            v2f a;
            a.x = th[m * LDSPITCH + kb];
            a.y = th[m * LDSPITCH + kb + 1];
            v2f b;
            b.x = B[(size_t)kb * DD + fcol];
            b.y = B[(size_t)(kb + 1) * DD + fcol];
            acc = __builtin_amdgcn_wmma_f32_16x16x4_f32(
                false, a, false, b, (short)0, acc, false, false);
        }

        // D layout: VGPR v -> M = v + rowh, N = fcol
        #pragma unroll
        for (int v = 0; v < 8; ++v) {
            const int node = row0 + v + rowh;
            hrel[(size_t)node * (RR * DD) + rel * DD + fcol] = acc[v];
        }
    }
}

// ---------------------------------------------------------------------------
// Kernel 2: out[n,f] = sum_d h[n,d]*root[d,f] + sum_d x[n,d]*skipW[f,d]
//                      + bias[f] + skip_bias[f]
// One block = one 16-node tile; 8 waves = 8 f-tiles of 16 (exactly D=128).
// ---------------------------------------------------------------------------
__global__ void __launch_bounds__(256)
k_base_wmma(const float* __restrict__ x,
            const float* __restrict__ h,
            const float* __restrict__ rootw,  // [D, D] (d, f)
            const float* __restrict__ bias,   // [D]
            const float* __restrict__ skipw,  // [D, D] (f, d)  (used transposed)
            const float* __restrict__ skipb,  // [D]
            float* __restrict__ out)          // [N, D]
{
    __shared__ float th[16 * LDSPITCH];
    __shared__ float tx[16 * LDSPITCH];

    const int row0 = blockIdx.x * 16;

    for (int i = threadIdx.x; i < 16 * 128; i += 256) {
        const int r = i >> 7;
        const int c = i & 127;
        th[r * LDSPITCH + c] = h[(size_t)(row0 + r) * DD + c];
        tx[r * LDSPITCH + c] = x[(size_t)(row0 + r) * DD + c];
    }
    __syncthreads();

    const int wave  = threadIdx.x >> 5;       // f tile 0..7
    const int lane  = threadIdx.x & 31;
    const int m     = lane & 15;
    const int khalf = (lane >> 4) << 1;
    const int rowh  = (lane >> 4) << 3;
    const int fcol  = wave * 16 + m;

    v8f acc = {};

    // Pass 1: h @ root_weight   (B[k][f] = rootw[k*D + f], strided loads)
    #pragma unroll 4
    for (int s = 0; s < 32; ++s) {
        const int kb = s * 4 + khalf;
        v2f a;
        a.x = th[m * LDSPITCH + kb];
        a.y = th[m * LDSPITCH + kb + 1];
        v2f b;
        b.x = rootw[(size_t)kb * DD + fcol];
        b.y = rootw[(size_t)(kb + 1) * DD + fcol];
        acc = __builtin_amdgcn_wmma_f32_16x16x4_f32(
            false, a, false, b, (short)0, acc, false, false);
    }

    // Pass 2: x @ skip_weight^T (B[k][f] = skipw[f*D + k], contiguous pair)
    #pragma unroll 4
    for (int s = 0; s < 32; ++s) {
        const int kb = s * 4 + khalf;
        v2f a;
        a.x = tx[m * LDSPITCH + kb];
        a.y = tx[m * LDSPITCH + kb + 1];
        v2f b;
        b.x = skipw[(size_t)fcol * DD + kb];
        b.y = skipw[(size_t)fcol * DD + kb + 1];
        acc = __builtin_amdgcn_wmma_f32_16x16x4_f32(
            false, a, false, b, (short)0, acc, false, false);
    }

    const float badd = bias[fcol] + skipb[fcol];

    #pragma unroll
    for (int v = 0; v < 8; ++v) {
        const int node = row0 + v + rowh;
        out[(size_t)node * DD + fcol] = acc[v] + badd;
    }
}

// ---------------------------------------------------------------------------
// Kernel 3: per-edge gather/scale/scatter-add.
// One wave per edge: 32 lanes x float4 = 128 contiguous floats gathered from
// h_rel[src, type], scaled by edge_weight, atomically added into out[dst].
// ---------------------------------------------------------------------------
__global__ void __launch_bounds__(256)
k_edge_scatter(const int*   __restrict__ ei,    // [2, E] flat: src=ei[e], dst=ei[E+e]
               const int*   __restrict__ et,    // [E]
               const float* __restrict__ ew,    // [E]
               const float* __restrict__ hrel,  // [N, R, D]
               float* __restrict__ out)         // [N, D]
{
    const int sub  = threadIdx.x >> 5;   // edge slot within block (8 per block)
    const int lane = threadIdx.x & 31;
    const int e    = blockIdx.x * 8 + sub;
    if (e >= EE) return;

    const int   src = ei[e];
    const int   dst = ei[EE + e];
    const int   t   = et[e];
    const float w   = ew[e];

    const float4* __restrict__ src4 =
        (const float4*)(hrel + (size_t)src * (RR * DD) + (size_t)t * DD);
    const float4 v = src4[lane];

    float* __restrict__ o = out + (size_t)dst * DD + lane * 4;
    atomicAdd(o + 0, v.x * w);
    atomicAdd(o + 1, v.y * w);
    atomicAdd(o + 2, v.z * w);
    atomicAdd(o + 3, v.w * w);
}

// ---------------------------------------------------------------------------
extern "C" void kernel_launch(void* const* d_in, const int* in_sizes, int n_in,
                              void* d_out, int out_size, void* d_ws, size_t ws_size,
                              hipStream_t stream)
{
    (void)in_sizes; (void)n_in; (void)out_size; (void)ws_size;

    const float* x      = (const float*)d_in[0];  // [N, D]
    const float* h      = (const float*)d_in[1];  // [N, D]
    const int*   ei     = (const int*)  d_in[2];  // [2, E]
    const int*   et     = (const int*)  d_in[3];  // [E]
    const float* ew     = (const float*)d_in[4];  // [E]
    const float* relw   = (const float*)d_in[5];  // [R, D, D]
    const float* rootw  = (const float*)d_in[6];  // [D, D]
    const float* bias   = (const float*)d_in[7];  // [D]
    const float* skipw  = (const float*)d_in[8];  // [D, D]
    const float* skipb  = (const float*)d_in[9];  // [D]

    float* out  = (float*)d_out;                  // [N, D]
    float* hrel = (float*)d_ws;                   // [N, R, D] = 204.8 MB scratch

    // 1) h_rel = per-relation GEMM (WMMA)
    k_hrel_wmma<<<dim3(NTIL), dim3(256), 0, stream>>>(h, relw, hrel);

    // 2) out = h@root + x@skip^T + bias + skip_bias (WMMA)
    k_base_wmma<<<dim3(NTIL), dim3(256), 0, stream>>>(x, h, rootw, bias, skipw, skipb, out);

    // 3) out += scatter-add of edge messages
    k_edge_scatter<<<dim3(EE / 8), dim3(256), 0, stream>>>(ei, et, ew, hrel, out);
}